// RNNAgent_3951369912519
// MI455X (gfx1250) — compile-verified
//
#include <hip/hip_runtime.h>
#include <math.h>

#define RTOT  32768   // BS * N
#define TILES 2048    // RTOT / 16  (one 16-row tile == one attention batch)

typedef __attribute__((ext_vector_type(2))) float v2f;
typedef __attribute__((ext_vector_type(8))) float v8f;

// ---------------- WMMA f32 16x16x4 fragment helpers ----------------
// A (16xK, row-major in LDS): lanes 0-15 -> K={kb,kb+1}, lanes 16-31 -> K={kb+2,kb+3}
__device__ __forceinline__ v2f ldsAfrag(const float* A, int lda, int kb, int lane) {
  int row = lane & 15;
  int k0  = kb + ((lane >> 4) << 1);
  v2f a; a.x = A[row * lda + k0]; a.y = A[row * lda + k0 + 1]; return a;
}
// B[k][n] = W[n][k] for C = A @ W^T with torch-layout W [Nout, K] in global
__device__ __forceinline__ v2f glbBfrag(const float* W, int ldk, int nb, int kb, int lane) {
  int n  = nb + (lane & 15);
  int k0 = kb + ((lane >> 4) << 1);
  const float* p = W + (size_t)n * ldk + k0;
  v2f b; b.x = p[0]; b.y = p[1]; return b;
}
__device__ __forceinline__ v2f glbBfragClamp(const float* W, int ldk, int nb, int kb,
                                             int nmax, int lane) {
  int n  = nb + (lane & 15); n = n < nmax ? n : nmax - 1;   // avoid OOB reads (fc2: 30 rows)
  int k0 = kb + ((lane >> 4) << 1);
  const float* p = W + (size_t)n * ldk + k0;
  v2f b; b.x = p[0]; b.y = p[1]; return b;
}
// B (KxN row-major in LDS)
__device__ __forceinline__ v2f ldsBfrag(const float* B, int ldn, int nb, int kb, int lane) {
  int n  = nb + (lane & 15);
  int k0 = kb + ((lane >> 4) << 1);
  v2f b; b.x = B[k0 * ldn + n]; b.y = B[(k0 + 1) * ldn + n]; return b;
}
__device__ __forceinline__ v8f wmma4(v2f a, v2f b, v8f c) {
  return __builtin_amdgcn_wmma_f32_16x16x4_f32(false, a, false, b, (short)0, c,
                                               false, false);
}
// C += A(16xK from LDS) * W^T(n-block nb), W columns offset by wk0 (for concat inputs)
__device__ __forceinline__ v8f gemm(const float* Alds, int lda, const float* W, int ldk,
                                    int wk0, int nb, int K, int lane, v8f acc) {
  for (int k = 0; k < K; k += 4) {
    v2f a = ldsAfrag(Alds, lda, k, lane);
    v2f b = glbBfrag(W, ldk, nb, wk0 + k, lane);
    acc = wmma4(a, b, acc);
  }
  return acc;
}
// C/D layout: vgpr v -> row v (lanes 0-15) / v+8 (lanes 16-31), col = nb + lane&15
__device__ __forceinline__ void storeC(float* D, int ldd, int nb, v8f c, int lane) {
  int col = nb + (lane & 15);
  int rb  = (lane >> 4) << 3;
  #pragma unroll
  for (int v = 0; v < 8; ++v) D[(rb + v) * ldd + col] = c[v];
}
__device__ __forceinline__ void loadTile(float* dst, const float* src, int nfloat, int lane) {
  const float4* s = (const float4*)src;
  float4* d = (float4*)dst;
  for (int i = lane; i < (nfloat >> 2); i += 32) d[i] = s[i];
}
__device__ __forceinline__ float lrelu(float x) { return x > 0.f ? x : 0.01f * x; }
__device__ __forceinline__ float sigm(float x)  { return 1.f / (1.f + __expf(-x)); }

// ============ Stage A: fc1+relu, GRU, latent encoder, me1 (pre-BN) ============
__global__ void __launch_bounds__(32)
stageA(const float* __restrict__ inputs, const float* __restrict__ hidden,
       const float* __restrict__ latent,
       const float* __restrict__ fc1_w, const float* __restrict__ fc1_b,
       const float* __restrict__ w_ih,  const float* __restrict__ w_hh,
       const float* __restrict__ b_ih,  const float* __restrict__ b_hh,
       const float* __restrict__ lse_w1, const float* __restrict__ lse_b1,
       const float* __restrict__ lse_w2, const float* __restrict__ lse_b2,
       const float* __restrict__ me_w1,  const float* __restrict__ me_b1,
       float* __restrict__ gH, float* __restrict__ gLe, float* __restrict__ gT1) {
  __shared__ __align__(16) float sIn[16 * 128];
  __shared__ __align__(16) float sX [16 * 128];
  __shared__ __align__(16) float sH [16 * 128];
  __shared__ __align__(16) float sHn[16 * 128];
  __shared__ __align__(16) float sLs[16 * 64];
  __shared__ __align__(16) float sL1[16 * 128];
  __shared__ __align__(16) float sLe[16 * 128];
  const int lane = threadIdx.x;
  const int r0   = blockIdx.x * 16;

  loadTile(sIn, inputs + (size_t)r0 * 128, 16 * 128, lane);
  loadTile(sH,  hidden + (size_t)r0 * 128, 16 * 128, lane);
  loadTile(sLs, latent + (size_t)r0 * 64,  16 * 64,  lane);
  __syncthreads();

  // x = relu(inputs @ fc1_w^T + b)
  for (int nc = 0; nc < 8; ++nc) {
    v8f acc = {};
    acc = gemm(sIn, 128, fc1_w, 128, 0, nc * 16, 128, lane, acc);
    float b = fc1_b[nc * 16 + (lane & 15)];
    #pragma unroll
    for (int v = 0; v < 8; ++v) { float y = acc[v] + b; acc[v] = y > 0.f ? y : 0.f; }
    storeC(sX, 128, nc * 16, acc, lane);
  }
  __syncthreads();

  // GRU cell: six fused WMMA accumulators per 16-col chunk, gates on fragments
  for (int nc = 0; nc < 8; ++nc) {
    v8f air = {}, aiz = {}, ain = {}, ahr = {}, ahz = {}, ahn = {};
    for (int k = 0; k < 128; k += 4) {
      v2f ax = ldsAfrag(sX, 128, k, lane);
      v2f ah = ldsAfrag(sH, 128, k, lane);
      v2f br = glbBfrag(w_ih,             128, nc * 16, k, lane);
      v2f bz = glbBfrag(w_ih + 128 * 128, 128, nc * 16, k, lane);
      v2f bn = glbBfrag(w_ih + 256 * 128, 128, nc * 16, k, lane);
      v2f cr = glbBfrag(w_hh,             128, nc * 16, k, lane);
      v2f cz = glbBfrag(w_hh + 128 * 128, 128, nc * 16, k, lane);
      v2f cn = glbBfrag(w_hh + 256 * 128, 128, nc * 16, k, lane);
      air = wmma4(ax, br, air); aiz = wmma4(ax, bz, aiz); ain = wmma4(ax, bn, ain);
      ahr = wmma4(ah, cr, ahr); ahz = wmma4(ah, cz, ahz); ahn = wmma4(ah, cn, ahn);
    }
    int col = nc * 16 + (lane & 15);
    int rb  = (lane >> 4) << 3;
    float bir = b_ih[col], biz = b_ih[128 + col], bin = b_ih[256 + col];
    float bhr = b_hh[col], bhz = b_hh[128 + col], bhn = b_hh[256 + col];
    #pragma unroll
    for (int v = 0; v < 8; ++v) {
      float r  = sigm((air[v] + bir) + (ahr[v] + bhr));
      float z  = sigm((aiz[v] + biz) + (ahz[v] + bhz));
      float ng = tanhf((ain[v] + bin) + r * (ahn[v] + bhn));
      float hp = sH[(rb + v) * 128 + col];
      float h  = ng + z * (hp - ng);
      sHn[(rb + v) * 128 + col] = h;
      gH[(size_t)(r0 + rb + v) * 128 + col] = h;
    }
  }
  __syncthreads();

  // latent_emb = lse2(relu(lse1(latent)))
  for (int nc = 0; nc < 8; ++nc) {
    v8f acc = {};
    acc = gemm(sLs, 64, lse_w1, 64, 0, nc * 16, 64, lane, acc);
    float b = lse_b1[nc * 16 + (lane & 15)];
    #pragma unroll
    for (int v = 0; v < 8; ++v) { float y = acc[v] + b; acc[v] = y > 0.f ? y : 0.f; }
    storeC(sL1, 128, nc * 16, acc, lane);
  }
  __syncthreads();
  for (int nc = 0; nc < 8; ++nc) {
    v8f acc = {};
    acc = gemm(sL1, 128, lse_w2, 128, 0, nc * 16, 128, lane, acc);
    int col = nc * 16 + (lane & 15);
    float b = lse_b2[col];
    int rb = (lane >> 4) << 3;
    #pragma unroll
    for (int v = 0; v < 8; ++v) acc[v] += b;
    storeC(sLe, 128, nc * 16, acc, lane);
    #pragma unroll
    for (int v = 0; v < 8; ++v) gLe[(size_t)(r0 + rb + v) * 128 + col] = acc[v];
  }
  __syncthreads();

  // t1 = [h, latent_emb] @ me_w1^T + b  (pre-BatchNorm, needed for batch stats)
  for (int nc = 0; nc < 16; ++nc) {
    v8f acc = {};
    acc = gemm(sHn, 128, me_w1, 256, 0,   nc * 16, 128, lane, acc);
    acc = gemm(sLe, 128, me_w1, 256, 128, nc * 16, 128, lane, acc);
    int col = nc * 16 + (lane & 15);
    float b = me_b1[col];
    int rb = (lane >> 4) << 3;
    #pragma unroll
    for (int v = 0; v < 8; ++v) gT1[(size_t)(r0 + rb + v) * 256 + col] = acc[v] + b;
  }
}

// ============ Stage B: deterministic BatchNorm batch statistics ============
__global__ void __launch_bounds__(256)
bnStats(const float* __restrict__ gT1, const float* __restrict__ g,
        const float* __restrict__ beta, float* __restrict__ scale,
        float* __restrict__ shift) {
  const int col = blockIdx.x;     // 256 columns
  const int tid = threadIdx.x;
  float s = 0.f, s2 = 0.f;
  for (int r = tid; r < RTOT; r += 256) {
    float x = gT1[(size_t)r * 256 + col];
    s += x; s2 += x * x;
  }
  __shared__ float rs[256], rs2[256];
  rs[tid] = s; rs2[tid] = s2; __syncthreads();
  for (int o = 128; o > 0; o >>= 1) {
    if (tid < o) { rs[tid] += rs[tid + o]; rs2[tid] += rs2[tid + o]; }
    __syncthreads();
  }
  if (tid == 0) {
    float m    = rs[0] / (float)RTOT;
    float var  = rs2[0] / (float)RTOT - m * m;     // biased variance (ddof=0)
    float rstd = rsqrtf(var + 1e-5f);
    float sc   = g[col] * rstd;
    scale[col] = sc;
    shift[col] = beta[col] - m * sc;
  }
}

// ==== Stage C: BN+lrelu, me2->pm, value/key/query, attention, fused fc2 ====
__global__ void __launch_bounds__(32)
stageC(const float* __restrict__ eps,
       const float* __restrict__ me_w2, const float* __restrict__ me_b2,
       const float* __restrict__ k_w,  const float* __restrict__ k_b,
       const float* __restrict__ q_w,  const float* __restrict__ q_b,
       const float* __restrict__ v_w1, const float* __restrict__ v_b1,
       const float* __restrict__ v_w2, const float* __restrict__ v_b2,
       const float* __restrict__ fc2_w, const float* __restrict__ fc2_b,
       const float* __restrict__ gH, const float* __restrict__ gLe,
       const float* __restrict__ gT1, const float* __restrict__ gScale,
       const float* __restrict__ gShift, float* __restrict__ out) {
  __shared__ __align__(16) float sT [16 * 256];
  __shared__ __align__(16) float sH [16 * 128];
  __shared__ __align__(16) float sLe[16 * 128];
  __shared__ __align__(16) float sPm[16 * 64];
  __shared__ __align__(16) float sV1[16 * 256];
  __shared__ __align__(16) float sVal[16 * 64];
  __shared__ __align__(16) float sKh[16 * 32];
  __shared__ __align__(16) float sQh[16 * 32];
  __shared__ __align__(16) float sAl[16 * 16];
  __shared__ __align__(16) float sWv[16 * 32];
  const int lane = threadIdx.x;
  const int r0   = blockIdx.x * 16;

  loadTile(sH,  gH  + (size_t)r0 * 128, 16 * 128, lane);
  loadTile(sLe, gLe + (size_t)r0 * 128, 16 * 128, lane);
  for (int i = lane; i < 16 * 256; i += 32) {           // BN apply + leaky relu
    int col = i & 255;
    float x = gT1[(size_t)r0 * 256 + i];
    sT[i] = lrelu(x * gScale[col] + gShift[col]);
  }
  __syncthreads();

  // me2 -> clamped (mean,std), temp t2 lives in sV1's first half
  float* sT2 = sV1;
  for (int nc = 0; nc < 8; ++nc) {
    v8f acc = {};
    acc = gemm(sT, 256, me_w2, 256, 0, nc * 16, 256, lane, acc);
    float b = me_b2[nc * 16 + (lane & 15)];
    #pragma unroll
    for (int v = 0; v < 8; ++v) acc[v] = lrelu(acc[v] + b);
    storeC(sT2, 128, nc * 16, acc, lane);
  }
  __syncthreads();
  for (int i = lane; i < 16 * 64; i += 32) {            // pm = mean + std*eps
    int row = i >> 6, c = i & 63;
    float m = fminf(fmaxf(sT2[row * 128 + c], -1.f), 1.f);
    float s = fminf(fmaxf(sT2[row * 128 + 64 + c], 0.5f), 1.f);
    sPm[i] = m + s * eps[(size_t)(r0 + row) * 64 + c];
  }
  __syncthreads();

  // v1 = lrelu([h, pm] @ v_w1^T + b)  (overwrites t2 temp; only sH/sPm are read)
  for (int nc = 0; nc < 16; ++nc) {
    v8f acc = {};
    acc = gemm(sH,  128, v_w1, 192, 0,   nc * 16, 128, lane, acc);
    acc = gemm(sPm, 64,  v_w1, 192, 128, nc * 16, 64,  lane, acc);
    float b = v_b1[nc * 16 + (lane & 15)];
    #pragma unroll
    for (int v = 0; v < 8; ++v) acc[v] = lrelu(acc[v] + b);
    storeC(sV1, 256, nc * 16, acc, lane);
  }
  __syncthreads();
  for (int nc = 0; nc < 4; ++nc) {                      // value
    v8f acc = {};
    acc = gemm(sV1, 256, v_w2, 256, 0, nc * 16, 256, lane, acc);
    float b = v_b2[nc * 16 + (lane & 15)];
    #pragma unroll
    for (int v = 0; v < 8; ++v) acc[v] += b;
    storeC(sVal, 64, nc * 16, acc, lane);
  }
  for (int nc = 0; nc < 2; ++nc) {                      // raw key / query
    v8f ack = {};
    ack = gemm(sH, 128, k_w, 128, 0, nc * 16, 128, lane, ack);
    int col = nc * 16 + (lane & 15);
    float bk = k_b[col];
    #pragma unroll
    for (int v = 0; v < 8; ++v) ack[v] += bk;
    storeC(sKh, 32, nc * 16, ack, lane);
    v8f acq = {};
    acq = gemm(sPm, 64, q_w, 64, 0, nc * 16, 64, lane, acq);
    float bq = q_b[col];
    #pragma unroll
    for (int v = 0; v < 8; ++v) acq[v] += bq;
    storeC(sQh, 32, nc * 16, acq, lane);
  }
  __syncthreads();

  {                                                     // row-wise L2 normalize
    int row = lane & 15;
    float* buf = (lane < 16) ? sKh : sQh;
    float ss = 0.f;
    for (int a = 0; a < 32; ++a) { float x = buf[row * 32 + a]; ss += x * x; }
    float inv = 1.f / fmaxf(sqrtf(ss), 1e-8f);
    for (int a = 0; a < 32; ++a) buf[row * 32 + a] *= inv;
  }
  __syncthreads();
  for (int i = lane; i < 256; i += 32) {                // cosine logits
    int ri = i >> 4, rj = i & 15;
    float s = 0.f;
    for (int a = 0; a < 32; ++a) s += sKh[ri * 32 + a] * sQh[rj * 32 + a];
    sAl[i] = s;
  }
  __syncthreads();
  if (lane < 16) {                                      // softmax_j then kill diag
    int i = lane;
    float mx = -1e30f;
    for (int j = 0; j < 16; ++j) mx = fmaxf(mx, sAl[i * 16 + j]);
    float sum = 0.f;
    for (int j = 0; j < 16; ++j) { float e = __expf(sAl[i * 16 + j] - mx); sAl[i * 16 + j] = e; sum += e; }
    float inv = 1.f / sum;
    for (int j = 0; j < 16; ++j) sAl[i * 16 + j] *= inv;
    sAl[i * 16 + i] = 0.f;
  }
  __syncthreads();
  // Wv[j][o] = value[j,:] . fc2_w[o, 256 + 64j : 256 + 64j + 64]
  for (int i = lane; i < 16 * 32; i += 32) {
    int j = i >> 5, o = i & 31;
    float s = 0.f;
    if (o < 30) {
      const float* w = fc2_w + (size_t)o * 1280 + 256 + j * 64;
      const float* v = sVal + j * 64;
      for (int c = 0; c < 64; ++c) s += v[c] * w[c];
    }
    sWv[j * 32 + o] = s;
  }
  __syncthreads();
  // fc2: out = h@W[:, :128]^T + le@W[:,128:256]^T + alpha' @ Wv + b
  for (int nc = 0; nc < 2; ++nc) {
    v8f acc = {};
    for (int k = 0; k < 128; k += 4) {
      v2f a = ldsAfrag(sH, 128, k, lane);
      v2f b = glbBfragClamp(fc2_w, 1280, nc * 16, k, 30, lane);
      acc = wmma4(a, b, acc);
    }
    for (int k = 0; k < 128; k += 4) {
      v2f a = ldsAfrag(sLe, 128, k, lane);
      v2f b = glbBfragClamp(fc2_w, 1280, nc * 16, 128 + k, 30, lane);
      acc = wmma4(a, b, acc);
    }
    for (int k = 0; k < 16; k += 4) {
      v2f a = ldsAfrag(sAl, 16, k, lane);
      v2f b = ldsBfrag(sWv, 32, nc * 16, k, lane);
      acc = wmma4(a, b, acc);
    }
    int col = nc * 16 + (lane & 15);
    if (col < 30) {
      float b = fc2_b[col];
      int rb = (lane >> 4) << 3;
      #pragma unroll
      for (int v = 0; v < 8; ++v) out[(size_t)(r0 + rb + v) * 30 + col] = acc[v] + b;
    }
  }
}

extern "C" void kernel_launch(void* const* d_in, const int* in_sizes, int n_in,
                              void* d_out, int out_size, void* d_ws, size_t ws_size,
                              hipStream_t stream) {
  (void)in_sizes; (void)n_in; (void)out_size; (void)ws_size;
  const float* inputs = (const float*)d_in[0];
  const float* hidden = (const float*)d_in[1];
  const float* latent = (const float*)d_in[2];
  const float* eps    = (const float*)d_in[3];
  const float* fc1_w  = (const float*)d_in[4];
  const float* fc1_b  = (const float*)d_in[5];
  const float* w_ih   = (const float*)d_in[6];
  const float* w_hh   = (const float*)d_in[7];
  const float* b_ih   = (const float*)d_in[8];
  const float* b_hh   = (const float*)d_in[9];
  const float* lse_w1 = (const float*)d_in[10];
  const float* lse_b1 = (const float*)d_in[11];
  const float* lse_w2 = (const float*)d_in[12];
  const float* lse_b2 = (const float*)d_in[13];
  const float* me_w1  = (const float*)d_in[14];
  const float* me_b1  = (const float*)d_in[15];
  const float* me_g   = (const float*)d_in[16];
  const float* me_beta= (const float*)d_in[17];
  const float* me_w2  = (const float*)d_in[18];
  const float* me_b2  = (const float*)d_in[19];
  const float* k_w    = (const float*)d_in[20];
  const float* k_b    = (const float*)d_in[21];
  const float* q_w    = (const float*)d_in[22];
  const float* q_b    = (const float*)d_in[23];
  const float* v_w1   = (const float*)d_in[24];
  const float* v_b1   = (const float*)d_in[25];
  const float* v_w2   = (const float*)d_in[26];
  const float* v_b2   = (const float*)d_in[27];
  const float* fc2_w  = (const float*)d_in[28];
  const float* fc2_b  = (const float*)d_in[29];
  float* out = (float*)d_out;

  float* ws    = (float*)d_ws;
  float* gH    = ws;                               // R*128
  float* gLe   = gH  + (size_t)RTOT * 128;         // R*128
  float* gT1   = gLe + (size_t)RTOT * 128;         // R*256
  float* gScale= gT1 + (size_t)RTOT * 256;         // 256
  float* gShift= gScale + 256;                     // 256

  stageA<<<TILES, 32, 0, stream>>>(inputs, hidden, latent, fc1_w, fc1_b, w_ih, w_hh,
                                   b_ih, b_hh, lse_w1, lse_b1, lse_w2, lse_b2,
                                   me_w1, me_b1, gH, gLe, gT1);
  bnStats<<<256, 256, 0, stream>>>(gT1, me_g, me_beta, gScale, gShift);
  stageC<<<TILES, 32, 0, stream>>>(eps, me_w2, me_b2, k_w, k_b, q_w, q_b,
                                   v_w1, v_b1, v_w2, v_b2, fc2_w, fc2_b,
                                   gH, gLe, gT1, gScale, gShift, out);
}